// LabelTripletLoss_30270929502704
// MI455X (gfx1250) — compile-verified
//
#include <hip/hip_runtime.h>

typedef __attribute__((ext_vector_type(2))) float v2f;
typedef __attribute__((ext_vector_type(4))) float v4f;
typedef __attribute__((ext_vector_type(8))) float v8f;

#define N_  48
#define W_  16
#define X_  256
#define NW_ 768

// workspace layout (in floats)
#define WS_NRM  0
#define WS_DIST 768
#define WS_TL   (768 + N_*NW_)
#define WS_HARD (WS_TL + N_)

// ---------------------------------------------------------------------------
// Kernel 1: row norms for all 768 flat rows (b.b); anchor norms are nrm[m*16]
// ---------------------------------------------------------------------------
__global__ void norms_kernel(const float* __restrict__ emb, float* __restrict__ ws) {
  int idx = blockIdx.x * 256 + threadIdx.x;
  if (idx >= NW_) return;
  const float* p = emb + (size_t)idx * X_;
  float s = 0.f;
  for (int k = 0; k < X_; k += 4) {
    v4f v = *(const v4f*)(p + k);
    s += v.x * v.x + v.y * v.y + v.z * v.z + v.w * v.w;
  }
  ws[WS_NRM + idx] = s;
}

// ---------------------------------------------------------------------------
// Kernel 2: dist[48][768] = relu(|a|^2 - 2 a.b + |b|^2) via V_WMMA_F32_16X16X4
// One wave (32 lanes) per 16x16 output tile; 64 WMMA steps over K=256.
// ---------------------------------------------------------------------------
__global__ __launch_bounds__(32)
void gemm_dist_kernel(const float* __restrict__ emb, float* __restrict__ ws) {
  const int tn   = blockIdx.x;      // column tile 0..47
  const int tm   = blockIdx.y;      // row tile    0..2
  const int lane = threadIdx.x;     // 0..31 (wave32)
  const int half = lane >> 4;       // 0: K=0..1 side, 1: K=2..3 side
  const int l15  = lane & 15;

  // A row = anchor (w==0) of embedding row (tm*16 + l15)
  const float* ap = emb + (size_t)(tm * 16 + l15) * (W_ * X_);
  // B column n = flat row (tn*16 + l15); B[k][n] = flat[n][k] (contiguous in k)
  const float* bp = emb + (size_t)(tn * 16 + l15) * X_;

  v8f c = {0.f, 0.f, 0.f, 0.f, 0.f, 0.f, 0.f, 0.f};
  for (int k0 = 0; k0 < X_; k0 += 4) {
    const int kk = k0 + half * 2;
    v2f a = *(const v2f*)(ap + kk);   // A[M=l15][kk], A[M][kk+1]
    v2f b = *(const v2f*)(bp + kk);   // B[kk][N=l15], B[kk+1][N]
    c = __builtin_amdgcn_wmma_f32_16x16x4_f32(
        /*neg_a=*/false, a, /*neg_b=*/false, b,
        /*c_mod=*/(short)0, c, /*reuse_a=*/false, /*reuse_b=*/false);
  }

  const float* nrm = ws + WS_NRM;
  float bb = nrm[tn * 16 + l15];
  float* dist = ws + WS_DIST;
#pragma unroll
  for (int r = 0; r < 8; ++r) {
    int M = tm * 16 + r + half * 8;   // C layout: VGPR r -> M=r (lanes 0-15), M=r+8 (16-31)
    float aa = nrm[M * 16];           // anchor M == flat row M*16
    float d = aa - 2.0f * c[r] + bb;
    d = d > 0.f ? d : 0.f;
    dist[(size_t)M * NW_ + tn * 16 + l15] = d;
  }
}

// ---------------------------------------------------------------------------
// Kernel 3: per-anchor-row statistics. One 256-thread block per row.
// ---------------------------------------------------------------------------
__global__ __launch_bounds__(256)
void stats_kernel(const float* __restrict__ dist, const float* __restrict__ lab,
                  float* __restrict__ out, float* __restrict__ tlsum,
                  float* __restrict__ hardsum) {
  const int n = blockIdx.x;
  const int tid = threadIdx.x;

  __shared__ float s_d[NW_];
  __shared__ unsigned char s_neg[NW_];
  __shared__ float s_posd[W_];
  __shared__ int   s_npos;
  __shared__ float redf[256];
  __shared__ int   redi[256];
  __shared__ float s_minv;
  __shared__ int   s_istar;

  for (int j = tid; j < NW_; j += 256) {
    s_d[j]   = dist[(size_t)n * NW_ + j];
    s_neg[j] = 1;
  }
  __syncthreads();
  if (tid < W_) {
    float l = lab[n * W_ + tid];
    if (l > 0.f || tid == 0) s_neg[n * W_ + tid] = 0;   // Pext: positives + anchor
  }
  if (tid == 0) {
    int cnt = 0;
    for (int w = 0; w < W_; ++w)
      if (lab[n * W_ + w] > 0.f) { s_posd[cnt] = s_d[n * W_ + w]; ++cnt; }
    s_npos = cnt;
  }
  __syncthreads();
  const int npos = s_npos;

  // per-thread partials over columns j = tid, tid+256, tid+512 (ascending)
  float minv = 1e30f; int mini = NW_;
  float sum_neg = 0.f, tl = 0.f, hard = 0.f, accu = 0.f;
  for (int j = tid; j < NW_; j += 256) {
    if (s_neg[j]) {
      float d = s_d[j];
      if (d < minv) { minv = d; mini = j; }   // strict < keeps first occurrence
      sum_neg += d;
      for (int p = 0; p < npos; ++p) {
        float dp = s_posd[p];
        float t = dp - d + 1.0f;              // MARGIN = 1.0
        float tr = t > 0.f ? t : 0.f;
        tl += tr;
        if (tr > 1e-16f) hard += 1.f;
        if (d > dp) accu += 1.f;              // diff = d_neg - d_pos > 0
      }
    }
  }

  // (min, idx) reduction with smallest-index tie-break (lexicographic argmin)
  redf[tid] = minv; redi[tid] = mini;
  __syncthreads();
  for (int s = 128; s > 0; s >>= 1) {
    if (tid < s) {
      float v2 = redf[tid + s]; int i2 = redi[tid + s];
      if (v2 < redf[tid] || (v2 == redf[tid] && i2 < redi[tid])) {
        redf[tid] = v2; redi[tid] = i2;
      }
    }
    __syncthreads();
  }
  if (tid == 0) { s_minv = redf[0]; s_istar = redi[0]; }
  __syncthreads();
  const float min_neg = s_minv;
  const int   istar   = s_istar;

  auto block_sum = [&](float v) -> float {
    redf[tid] = v; __syncthreads();
    for (int s = 128; s > 0; s >>= 1) {
      if (tid < s) redf[tid] += redf[tid + s];
      __syncthreads();
    }
    float r = redf[0]; __syncthreads();
    return r;
  };

  // rank of istar among negatives = #negatives strictly before it
  float ncnt = 0.f;
  for (int j = tid; j < istar; j += 256) ncnt += s_neg[j] ? 1.f : 0.f;
  const float neg_rank      = block_sum(ncnt);
  const float sum_neg_total = block_sum(sum_neg);
  const float tl_total      = block_sum(tl);
  const float hard_total    = block_sum(hard);
  const float accu_total    = block_sum(accu);

  if (tid == 0) {
    float maxp = -1e30f; int pstar = 0; float sum_pos = 0.f;
    for (int p = 0; p < npos; ++p) {
      float d = s_posd[p];
      sum_pos += d;
      if (d > maxp) { maxp = d; pstar = p; }   // first occurrence == pos rank
    }
    float fnpos  = (float)npos;
    float fnneg  = (float)(NW_ - npos - 1);
    float npairs = fnpos * fnneg;
    out[  1 + n] = maxp;                                            // max_anchor2pos
    out[ 49 + n] = min_neg;                                         // min_anchor2neg
    out[ 97 + n] = min_neg - maxp;                                  // min_neg2pos
    out[145 + n] = (fnpos * sum_neg_total - fnneg * sum_pos) / npairs; // avg_neg2pos
    out[193 + n] = neg_rank;                                        // neg_idx
    out[241 + n] = (float)pstar;                                    // pos_idx
    out[289 + n] = accu_total / npairs;                             // accu_ratio
    tlsum[n]   = tl_total;
    hardsum[n] = hard_total;
  }
}

// ---------------------------------------------------------------------------
// Kernel 4: deterministic final scalar
// ---------------------------------------------------------------------------
__global__ void finalize_kernel(const float* __restrict__ tlsum,
                                const float* __restrict__ hardsum,
                                float* __restrict__ out) {
  if (threadIdx.x == 0 && blockIdx.x == 0) {
    float s = 0.f, h = 0.f;
    for (int i = 0; i < N_; ++i) { s += tlsum[i]; h += hardsum[i]; }
    out[0] = s / (h + 1e-16f);
  }
}

extern "C" void kernel_launch(void* const* d_in, const int* in_sizes, int n_in,
                              void* d_out, int out_size, void* d_ws, size_t ws_size,
                              hipStream_t stream) {
  const float* emb = (const float*)d_in[0];   // (48,16,256) f32
  const float* lab = (const float*)d_in[1];   // (48,16) f32
  float* out = (float*)d_out;                 // 337 f32
  float* ws  = (float*)d_ws;

  norms_kernel<<<3, 256, 0, stream>>>(emb, ws);
  dim3 grid(48, 3);
  gemm_dist_kernel<<<grid, 32, 0, stream>>>(emb, ws);
  stats_kernel<<<N_, 256, 0, stream>>>(ws + WS_DIST, lab, out,
                                       ws + WS_TL, ws + WS_HARD);
  finalize_kernel<<<1, 32, 0, stream>>>(ws + WS_TL, ws + WS_HARD, out);
}